// HybridRWKVMambaBlock_68642167325120
// MI455X (gfx1250) — compile-verified
//
#include <hip/hip_runtime.h>

// RWKV/Mamba selective scan on gfx1250 via chunked linear attention.
// B=4, T=2048, H=8, D=64. fp32 throughout (V_WMMA_F32_16X16X4_F32): the
// kernel is HBM-bound (~84MB -> ~3.6us at 23.3 TB/s) and the multiplicative
// scan is precision-sensitive, so bf16/fp8 WMMA would be the wrong trade.
//
// Grid: (B*H, 4); each block owns one (b,h) scan restricted to a 16-wide
// slice of the state's n-dimension (state columns are independent; the
// redundant k/w/r reads across the 4 n-groups are served by the 192MB L2 —
// the whole 84MB working set is L2-resident). Block = 256 threads = 8 waves.
//
// Chunk C=16 timesteps, with A_t[m] = prod_{s<=t}(1 - w_s[m]):
//   qt = r .* A_t ; khat = k ./ A_t ; kbar = k .* (A_C ./ A_t)
//   P  = tril(qt @ khat^T)          (16x16, K=64)    -> wave 0
//   Y  = qt @ S0 + P @ Vg           (16x16, K=64+16) -> wave 1 (stores y)
//   S  = diag(A_C).*S0 + kbar^T@Vg  (64x16)          -> waves 4..7
//
// Staging uses the CDNA5 async path: GLOBAL_LOAD_ASYNC_TO_LDS_B128 DMAs the
// next chunk HBM->LDS (double-buffered) under the current chunk's compute,
// tracked with ASYNCcnt and completed with s_wait_asynccnt.

#define TD 2048
#define HD 8
#define DD 64
#define CHUNK 16
#define NCHUNK (TD / CHUNK)   // 128
#define NGROUP 4
#define GW 16                 // n-columns per block
#define ROWSTRIDE (HD * DD)   // 512 floats between consecutive t
#define SK 68                 // LDS stride for 16x64 arrays (16B aligned, bank-skewed)
#define SV 20                 // LDS stride for 16x16 arrays
#define SS 20                 // LDS stride for 64x16 state

typedef __attribute__((ext_vector_type(2))) float v2f;
typedef __attribute__((ext_vector_type(8))) float v8f;

__device__ __forceinline__ v8f wmma_f32x4(v2f a, v2f b, v8f c) {
  // D = A(16x4,f32) * B(4x16,f32) + C(16x16,f32)
  return __builtin_amdgcn_wmma_f32_16x16x4_f32(
      /*neg_a=*/false, a, /*neg_b=*/false, b,
      /*c_mod=*/(short)0, c, /*reuse_a=*/false, /*reuse_b=*/false);
}

// LDS byte offset of a __shared__ object: the shared aperture places the
// LDS offset in flat-address bits [31:0] (ISA 10.2 aperture mapping).
__device__ __forceinline__ unsigned lds_off(const void* p) {
  return (unsigned)(unsigned long long)(uintptr_t)p;
}

// HBM -> LDS direct DMA, 16B per lane, ASYNCcnt-tracked (ISA 10. / 15.18.3).
// GVS addressing: mem = SGPR64 + i32 VGPR offset.
__device__ __forceinline__ void async_b128(unsigned lds_byte, unsigned gbyte_off,
                                           const float* sbase) {
  asm volatile("global_load_async_to_lds_b128 %0, %1, %2"
               :: "v"(lds_byte), "v"(gbyte_off), "s"(sbase)
               : "memory");
}

__device__ __forceinline__ void wait_async0() {
  asm volatile("s_wait_asynccnt 0x0" ::: "memory");
}

__global__ __launch_bounds__(256) void rwkv_scan_wmma(
    const float* __restrict__ gk, const float* __restrict__ gv,
    const float* __restrict__ gw, const float* __restrict__ gr,
    float* __restrict__ gy) {
  __shared__ float kst[2][CHUNK * SK], wst[2][CHUNK * SK], rst[2][CHUNK * SK];
  __shared__ float vst[2][CHUNK * SV];
  __shared__ float qt[CHUNK * SK], khat[CHUNK * SK], kbar[CHUNK * SK];
  __shared__ float Acum[CHUNK * SK];
  __shared__ float aCv[DD];
  __shared__ float Sst[DD * SS];     // carried state, this block's 16 n-columns
  __shared__ float Pst[CHUNK * SV];  // masked intra-chunk scores

  const int tid = threadIdx.x;
  const int lane = tid & 31;
  const int wave = tid >> 5;
  const int l15 = lane & 15;
  const int kh2 = (lane >> 4) * 2;   // A/B fragment K offset per lane half

  const int bh = blockIdx.x;
  const int b = bh >> 3, h = bh & 7;
  const int ng = blockIdx.y;

  const size_t baseBH = (size_t)b * (TD * HD * DD) + (size_t)h * DD;
  const float* gkB = gk + baseBH;  // uniform -> SGPR pair for async saddr
  const float* gvB = gv + baseBH;
  const float* gwB = gw + baseBH;
  const float* grB = gr + baseBH;

  // zero carried state
  for (int i = tid; i < DD * SS; i += 256) Sst[i] = 0.0f;

  // staging assignment: k/w/r full 16x64 rows, v only this block's 16 columns
  const int srow = tid >> 4;            // 0..15
  const int sc4 = (tid & 15) * 4;       // 0..60
  const int vrow = tid >> 2;            // 0..15 (tid < 64)
  const int vcol = (tid & 3) * 4;       // 0..12
  const int vgc = ng * GW + vcol;       // global n column

  // issue chunk-0 staging DMA
  {
    unsigned go = (unsigned)(((0 * CHUNK + srow) * ROWSTRIDE + sc4) * 4);
    async_b128(lds_off(&kst[0][srow * SK + sc4]), go, gkB);
    async_b128(lds_off(&wst[0][srow * SK + sc4]), go, gwB);
    async_b128(lds_off(&rst[0][srow * SK + sc4]), go, grB);
    if (tid < 64) {
      unsigned gov = (unsigned)(((0 * CHUNK + vrow) * ROWSTRIDE + vgc) * 4);
      async_b128(lds_off(&vst[0][vrow * SV + vcol]), gov, gvB);
    }
  }

  for (int c = 0; c < NCHUNK; ++c) {
    const int cur = c & 1;
    wait_async0();    // chunk-c staging DMA (this wave's) has landed in LDS
    __syncthreads();  // all waves landed; prev chunk fully consumed LDS

    if (c + 1 < NCHUNK) {  // DMA next chunk into the other buffer
      const int nxt = cur ^ 1;
      unsigned go = (unsigned)((((c + 1) * CHUNK + srow) * ROWSTRIDE + sc4) * 4);
      async_b128(lds_off(&kst[nxt][srow * SK + sc4]), go, gkB);
      async_b128(lds_off(&wst[nxt][srow * SK + sc4]), go, gwB);
      async_b128(lds_off(&rst[nxt][srow * SK + sc4]), go, grB);
      if (tid < 64) {
        unsigned gov = (unsigned)((((c + 1) * CHUNK + vrow) * ROWSTRIDE + vgc) * 4);
        async_b128(lds_off(&vst[nxt][vrow * SV + vcol]), gov, gvB);
      }
    }

    const float* kc = kst[cur];
    const float* wc = wst[cur];
    const float* rc = rst[cur];
    const float* vc = vst[cur];

    // per-feature cumulative decay products (64 lanes, 16 sequential steps)
    if (tid < DD) {
      float a = 1.0f;
#pragma unroll
      for (int t = 0; t < CHUNK; ++t) {
        a *= (1.0f - wc[t * SK + tid]);
        Acum[t * SK + tid] = a;
      }
      aCv[tid] = a;
    }
    __syncthreads();

    // derived operand matrices (1024 elems / 256 threads)
#pragma unroll
    for (int ii = 0; ii < 4; ++ii) {
      int idx = tid + 256 * ii;
      int t = idx >> 6, m = idx & 63;
      float a = Acum[t * SK + m];
      float kk = kc[t * SK + m];
      float kh_ = kk / a;                   // A_t in (0,1], never zero (w<1)
      qt[t * SK + m] = rc[t * SK + m] * a;
      khat[t * SK + m] = kh_;
      kbar[t * SK + m] = kh_ * aCv[m];      // = k * (A_C / A_t) <= |k|
    }
    __syncthreads();

    v8f yacc = {};
    if (wave == 0) {
      // P = tril(qt @ khat^T), one 16x16 tile, K = 64
      v8f acc = {};
#pragma unroll
      for (int q = 0; q < 16; ++q) {
        int m0 = q * 4 + kh2;
        v2f a, bb;
        a.x = qt[l15 * SK + m0];
        a.y = qt[l15 * SK + m0 + 1];
        bb.x = khat[l15 * SK + m0];  // B[K=m][N=s] = khat[s][m]
        bb.y = khat[l15 * SK + m0 + 1];
        acc = wmma_f32x4(a, bb, acc);
      }
#pragma unroll
      for (int g = 0; g < 8; ++g) {
        int t = g + 8 * (lane >> 4);
        Pst[t * SV + l15] = (l15 <= t) ? acc[g] : 0.0f;  // causal mask
      }
    } else if (wave == 1) {
      // Y1 = qt @ S0 (reads carried state), K = 64
#pragma unroll
      for (int q = 0; q < 16; ++q) {
        int m0 = q * 4 + kh2;
        v2f a, bb;
        a.x = qt[l15 * SK + m0];
        a.y = qt[l15 * SK + m0 + 1];
        bb.x = Sst[m0 * SS + l15];
        bb.y = Sst[(m0 + 1) * SS + l15];
        yacc = wmma_f32x4(a, bb, yacc);
      }
    }
    __syncthreads();  // P ready; wave1 done reading S0

    if (wave == 1) {
      // Y += P @ Vg, K = 16; then store output tile
#pragma unroll
      for (int q = 0; q < 4; ++q) {
        int s0 = q * 4 + kh2;
        v2f a, bb;
        a.x = Pst[l15 * SV + s0];
        a.y = Pst[l15 * SV + s0 + 1];
        bb.x = vc[s0 * SV + l15];
        bb.y = vc[(s0 + 1) * SV + l15];
        yacc = wmma_f32x4(a, bb, yacc);
      }
#pragma unroll
      for (int g = 0; g < 8; ++g) {
        int t = g + 8 * (lane >> 4);
        size_t off =
            baseBH + (size_t)(c * CHUNK + t) * ROWSTRIDE + ng * GW + l15;
        gy[off] = yacc[g];
      }
    } else if (wave >= 4) {
      // state update: S = diag(A_C).*S0 + kbar^T @ Vg, row tile i (16 rows)
      int i = wave - 4;
      v8f cs;
#pragma unroll
      for (int g = 0; g < 8; ++g) {
        int m = 16 * i + g + 8 * (lane >> 4);
        cs[g] = Sst[m * SS + l15] * aCv[m];
      }
#pragma unroll
      for (int q = 0; q < 4; ++q) {
        int s0 = q * 4 + kh2;
        v2f a, bb;
        a.x = kbar[s0 * SK + 16 * i + l15];  // A[M=m][K=t] = kbar[t][m]
        a.y = kbar[(s0 + 1) * SK + 16 * i + l15];
        bb.x = vc[s0 * SV + l15];
        bb.y = vc[(s0 + 1) * SV + l15];
        cs = wmma_f32x4(a, bb, cs);
      }
#pragma unroll
      for (int g = 0; g < 8; ++g) {
        int m = 16 * i + g + 8 * (lane >> 4);
        Sst[m * SS + l15] = cs[g];
      }
    }
  }
}

extern "C" void kernel_launch(void* const* d_in, const int* in_sizes, int n_in,
                              void* d_out, int out_size, void* d_ws,
                              size_t ws_size, hipStream_t stream) {
  (void)in_sizes; (void)n_in; (void)out_size; (void)d_ws; (void)ws_size;
  const float* k = (const float*)d_in[0];
  const float* v = (const float*)d_in[1];
  const float* w = (const float*)d_in[2];
  const float* r = (const float*)d_in[3];
  float* y = (float*)d_out;
  dim3 grid(4 * HD, NGROUP);  // (b*h, n-group)
  rwkv_scan_wmma<<<grid, 256, 0, stream>>>(k, v, w, r, y);
}